// FernSparseTable_37512244364036
// MI455X (gfx1250) — compile-verified
//
#include <hip/hip_runtime.h>

// Problem constants (from the reference)
#define N_     32
#define M_     16
#define K_     12
#define P_     8
#define D_     64
#define HW_    4096        // 64*64
#define TABLE_ 4096        // 2^K

#define WAVES_PER_BLOCK 4
#define THREADS (WAVES_PER_BLOCK * 32)
#define PAIR_BYTES_PER_WAVE (32 * 128 * 8)                         // 32 KB: [32 px][128 pairs] * 8B
#define XSTAGE_OFF (WAVES_PER_BLOCK * PAIR_BYTES_PER_WAVE)         // 128 KB
#define XSTAGE_PER_WAVE (2 * K_ * 32 * 4)                          // 3072 B (double buffer)
#define LDS_TOTAL (XSTAGE_OFF + WAVES_PER_BLOCK * XSTAGE_PER_WAVE) // 143360 B

struct Pair { float a; int idx; };

// CDNA5 async global -> LDS copies (per-lane addresses, tracked by ASYNCcnt)
__device__ __forceinline__ void async_ld_b128(unsigned lds_addr, const void* gaddr) {
    asm volatile("global_load_async_to_lds_b128 %0, %1, off"
                 :: "v"(lds_addr), "v"(gaddr) : "memory");
}

extern "C" __global__ __launch_bounds__(THREADS)
void fern_sparse_table_kernel(const float* __restrict__ x,
                              const float* __restrict__ wtab,
                              float* __restrict__ out) {
    extern __shared__ char smem[];
    const int lane = threadIdx.x & 31;
    const int wave = threadIdx.x >> 5;
    const int pb   = blockIdx.x * WAVES_PER_BLOCK + wave;  // pixel-block (32 px each)
    const int n    = pb >> 7;                              // 128 pixel-blocks per image
    const int hw0  = (pb & 127) * 32;

    Pair* pairs = (Pair*)(smem + wave * PAIR_BYTES_PER_WAVE);      // [32 px][128 pairs]
    char* xs_g  = smem + XSTAGE_OFF + wave * XSTAGE_PER_WAVE;
    const unsigned xs_lds = (unsigned)(unsigned long long)(void*)xs_g;

    // Per-lane 16B chunk mapping for one fern's 12 channel planes (1536B = 3 x b128/lane)
    // chunk c (= i*32+lane): plane k = c>>3, intra-plane float offset = (c&7)*4
    // LDS byte = c*16  -> plane k occupies [k*128, k*128+128) == xb[k*32 .. k*32+31]
    const size_t xpix = (size_t)n * (M_ * K_) * HW_ + (size_t)hw0;

    // ---------------- Phase 1: bits -> (activation, table-index) pairs ----------------
    // Preload fern 0 (buffer 0)
    #pragma unroll
    for (int i = 0; i < 3; ++i) {
        const int c = i * 32 + lane;
        async_ld_b128(xs_lds + (unsigned)(c * 16),
                      x + xpix + (size_t)(c >> 3) * HW_ + (c & 7) * 4);
    }

    for (int m = 0; m < M_; ++m) {
        if (m + 1 < M_) {
            // prior LDS reads of the target buffer must drain before async writes land
            asm volatile("s_wait_dscnt 0" ::: "memory");
            const size_t  xf = xpix + (size_t)(m + 1) * K_ * HW_;
            const unsigned l2 = xs_lds + (unsigned)(((m + 1) & 1) * 1536);
            #pragma unroll
            for (int i = 0; i < 3; ++i) {
                const int c = i * 32 + lane;
                async_ld_b128(l2 + (unsigned)(c * 16),
                              x + xf + (size_t)(c >> 3) * HW_ + (c & 7) * 4);
            }
            asm volatile("s_wait_asynccnt 3" ::: "memory");  // fern m's 3 loads done
        } else {
            asm volatile("s_wait_asynccnt 0" ::: "memory");
        }

        const float* xb = (const float*)(xs_g + (m & 1) * 1536);
        float T[K_], BA[K_];
        int WB = 0;
        #pragma unroll
        for (int k = 0; k < K_; ++k) {
            const float t = xb[k * 32 + lane];
            T[k]  = t;
            WB   |= ((int)rintf(t)) << (11 - k);   // round-half-even == jnp.round
            BA[k] = fabsf(t - 0.5f);
        }

        // LP=3 most ambiguous bits: iterative first-argmin with exclusion
        int ai[3]; float av[3];
        #pragma unroll
        for (int j = 0; j < 3; ++j) {
            float mv = BA[0], vv = T[0]; int mi = 0;
            #pragma unroll
            for (int k = 1; k < K_; ++k) {
                const bool lt = BA[k] < mv;
                mv = lt ? BA[k] : mv;
                vv = lt ? T[k]  : vv;
                mi = lt ? k     : mi;
            }
            ai[j] = mi; av[j] = vv;
            #pragma unroll
            for (int k = 0; k < K_; ++k) BA[k] += (k == mi) ? 1.0f : 0.0f;
        }

        const int m0 = 1 << (11 - ai[0]);
        const int m1 = 1 << (11 - ai[1]);
        const int m2 = 1 << (11 - ai[2]);
        const int base = WB & ~(m0 | m1 | m2);

        // rotate-swizzled pair slots: bankpair = (i + lane) % 32 -> conflict-free b64 stores
        Pair* pblock = pairs + lane * 128;
        #pragma unroll
        for (int p = 0; p < P_; ++p) {
            const float t0 = (p & 1) ? av[0] : 1.0f - av[0];
            const float t1 = (p & 2) ? av[1] : 1.0f - av[1];
            const float t2 = (p & 4) ? av[2] : 1.0f - av[2];
            Pair pr;
            pr.a   = t0 * t1 * t2;
            pr.idx = base | ((p & 1) ? m0 : 0) | ((p & 2) ? m1 : 0) | ((p & 4) ? m2 : 0);
            pblock[(m * 8 + p + lane) & 127] = pr;
        }
    }
    __syncthreads();

    // ---------------- Phase 2: gather-weighted accumulation, wave-wide over D ----------------
    // Lane L accumulates d = 2L, 2L+1; each table row read = one coalesced 256B b64 load.
    // Fern loop unrolled x2 -> 16 independent row gathers in flight to cover L2 latency.
    float* fwave = (float*)pairs;                  // out-tile overlay on consumed pair blocks
    const float* wl = wtab + 2 * lane;
    #pragma unroll 1
    for (int q = 0; q < 32; ++q) {
        float acc0 = 0.0f, acc1 = 0.0f;
        const Pair* pq = pairs + q * 128;
        #pragma unroll 2
        for (int m = 0; m < M_; ++m) {
            const float* wm = wl + (size_t)m * (TABLE_ * D_);
            #pragma unroll
            for (int p = 0; p < P_; ++p) {
                const Pair pr = pq[(m * 8 + p + q) & 127];   // uniform addr -> ds broadcast
                const float2 r = *(const float2*)(wm + (unsigned)pr.idx * D_);
                acc0 = fmaf(pr.a, r.x, acc0);
                acc1 = fmaf(pr.a, r.y, acc1);
            }
        }
        // park results in LDS (overlay on pixel q's consumed pair block), swizzled
        fwave[q * 256 + ((2 * lane + q) & 63)]     = acc0;
        fwave[q * 256 + ((2 * lane + 1 + q) & 63)] = acc1;
    }

    // ---------------- Phase 3: coalesced output stores via LDS transpose ----------------
    float* op = out + (size_t)n * D_ * HW_ + hw0 + lane;
    #pragma unroll 1
    for (int d = 0; d < D_; ++d) {
        op[(size_t)d * HW_] = fwave[lane * 256 + ((d + lane) & 63)];
    }
}

extern "C" void kernel_launch(void* const* d_in, const int* in_sizes, int n_in,
                              void* d_out, int out_size, void* d_ws, size_t ws_size,
                              hipStream_t stream) {
    const float* x = (const float*)d_in[0];   // (N, M*K, H, W) fp32
    const float* w = (const float*)d_in[1];   // (M, 4096, D)  fp32
    float* out = (float*)d_out;               // (N, D, H, W)  fp32

    const int pixel_blocks = N_ * (HW_ / 32);            // 4096 waves of 32 px
    const int grid = pixel_blocks / WAVES_PER_BLOCK;     // 1024 blocks, 128 thr each
    fern_sparse_table_kernel<<<grid, THREADS, LDS_TOTAL, stream>>>(x, w, out);
}